// YOLOLoss_74620761801586
// MI455X (gfx1250) — compile-verified
//
#include <hip/hip_runtime.h>
#include <hip/hip_bf16.h>
#include <stdint.h>

#define LAMBDA_COORD 5.0f
#define LAMBDA_NOOBJ 0.5f
#define CELL_FLOATS  30
#define BLK          256
#define WAVES        (BLK / 32)
#define CHUNK_FLOATS (BLK * CELL_FLOATS)   // 7680 floats per tensor per block
#define CHUNK_VECS   (CHUNK_FLOATS / 4)    // 1920 b128 transfers per tensor

// CDNA5 async global->LDS 128-bit copy (gfx1250). Per-lane: writes 16 bytes
// at the LDS byte offset in %0 from the 64-bit global address in %1.
// Tracked with ASYNCcnt.
__device__ __forceinline__ void async_g2l_b128(uint32_t lds_off, const float* g) {
    asm volatile("global_load_async_to_lds_b128 %0, %1, off"
                 :: "v"(lds_off), "v"((uint64_t)(uintptr_t)g)
                 : "memory");
}

__device__ __forceinline__ void wait_async_zero() {
    asm volatile("s_wait_asynccnt 0" ::: "memory");
}

__device__ __forceinline__ float iou_box(float ax, float ay, float aw, float ah,
                                         float bx, float by, float bw, float bh) {
    float al = ax - aw * 0.5f, ar = ax + aw * 0.5f;
    float at = ay - ah * 0.5f, ab = ay + ah * 0.5f;
    float bl = bx - bw * 0.5f, br = bx + bw * 0.5f;
    float bt = by - bh * 0.5f, bb = by + bh * 0.5f;
    float iw = fmaxf(fminf(ar, br) - fmaxf(al, bl), 0.0f);
    float ih = fmaxf(fminf(ab, bb) - fmaxf(at, bt), 0.0f);
    float inter = iw * ih;
    float uni = aw * ah + bw * bh - inter + 1e-10f;
    return inter / uni;
}

// wave32 full reduction via lane shuffles (no LDS, no barriers)
__device__ __forceinline__ float wave_reduce_add(float v) {
    #pragma unroll
    for (int m = 16; m >= 1; m >>= 1) v += __shfl_xor(v, m, 32);
    return v;
}

__global__ __launch_bounds__(BLK)
void yolo_loss_partial(const float* __restrict__ preds,
                       const float* __restrict__ labels,
                       float* __restrict__ partials,
                       int ncells) {
    __shared__ float sP[CHUNK_FLOATS];
    __shared__ float sL[CHUNK_FLOATS];
    __shared__ float wsum[WAVES];

    const int tid   = threadIdx.x;
    const int cell0 = blockIdx.x * BLK;
    const size_t base = (size_t)cell0 * CELL_FLOATS;

    // Number of valid b128 vectors this block may stream (guard the tail).
    const int validCells = min(BLK, ncells - cell0);
    const int validVecs  = (validCells * CELL_FLOATS) / 4;

    const uint32_t sPbase = (uint32_t)(uintptr_t)&sP[0];
    const uint32_t sLbase = (uint32_t)(uintptr_t)&sL[0];

    // Stage 61.4 KB (both tensors) into LDS with CDNA5 async 128-bit copies.
    for (int v = tid; v < validVecs; v += BLK) {
        const uint32_t off = (uint32_t)(v << 4);
        async_g2l_b128(sPbase + off, preds  + base + (size_t)v * 4);
        async_g2l_b128(sLbase + off, labels + base + (size_t)v * 4);
    }
    wait_async_zero();     // this wave's async transfers complete
    __syncthreads();       // all waves' transfers visible in LDS

    float loss = 0.0f;
    if (tid < validCells) {
        // record base tid*30 floats: even index -> 8-byte aligned -> float2 loads
        const float2* p2 = reinterpret_cast<const float2*>(&sP[tid * CELL_FLOATS]);
        const float2* l2 = reinterpret_cast<const float2*>(&sL[tid * CELL_FLOATS]);

        float2 pA = p2[0];                 // p0 p1
        float2 pB = p2[1];                 // p2 p3
        float2 pC = p2[2];                 // pc q0
        float2 pD = p2[3];                 // q1 q2
        float2 pE = p2[4];                 // q3 qc
        float2 gA = l2[0];                 // g0 g1
        float2 gB = l2[1];                 // g2 g3
        float2 gC = l2[2];                 // obj .

        float p0 = pA.x, p1 = pA.y, p2v = pB.x, p3 = pB.y, pc = pC.x;
        float q0 = pC.y, q1 = pD.x, q2v = pD.y, q3 = pE.x, qc = pE.y;
        float g0 = gA.x, g1 = gA.y, g2v = gB.x, g3 = gB.y;
        float obj   = gC.x;
        float noobj = 1.0f - obj;

        float iou1 = iou_box(p0, p1, p2v, p3, g0, g1, g2v, g3);
        float iou2 = iou_box(q0, q1, q2v, q3, g0, g1, g2v, g3);
        bool  r    = iou1 > iou2;

        float rb0 = r ? p0 : q0;
        float rb1 = r ? p1 : q1;
        float rb2 = r ? p2v : q2v;
        float rb3 = r ? p3 : q3;
        float rconf = r ? pc : qc;
        float oconf = r ? qc : pc;
        float riou  = fmaxf(iou1, iou2);

        // coord (xy)
        float dx = rb0 - g0, dy = rb1 - g1;
        float lxy = dx * dx + dy * dy;

        // coord (sqrt wh)
        float sw = sqrtf(fmaxf(rb2, 1e-8f)) - sqrtf(fmaxf(g2v, 1e-8f));
        float sh = sqrtf(fmaxf(rb3, 1e-8f)) - sqrtf(fmaxf(g3, 1e-8f));
        float lwh = sw * sw + sh * sh;

        // objectness
        float dob = rconf - riou;
        float lobj = dob * dob;

        // no-object + non-responsible conf
        float lno = noobj * (pc * pc + qc * qc) + obj * (oconf * oconf);

        // 20-class SSE, float2-vectorized (offset tid*30+10 is even -> aligned)
        float lcls = 0.0f;
        #pragma unroll
        for (int k = 5; k < 15; ++k) {      // float2 indices 5..14 == floats 10..29
            float2 pv = p2[k];
            float2 lv = l2[k];
            float d0 = pv.x - lv.x;
            float d1 = pv.y - lv.y;
            lcls += d0 * d0 + d1 * d1;
        }

        loss = LAMBDA_COORD * obj * (lxy + lwh)
             + obj * lobj
             + LAMBDA_NOOBJ * lno
             + obj * lcls;
    }

    // wave32 shuffle reduction, then one LDS combine of the 8 wave sums
    loss = wave_reduce_add(loss);
    const int wave = tid >> 5, lane = tid & 31;
    if (lane == 0) wsum[wave] = loss;
    __syncthreads();
    if (tid < WAVES) {
        float s = wsum[tid];
        #pragma unroll
        for (int m = WAVES / 2; m >= 1; m >>= 1) s += __shfl_xor(s, m, 32);
        if (tid == 0) partials[blockIdx.x] = s;
    }
}

__global__ __launch_bounds__(BLK)
void yolo_loss_final(const float* __restrict__ partials, int n,
                     float* __restrict__ out, float inv_batch) {
    __shared__ float wsum[WAVES];
    float s = 0.0f;
    for (int i = threadIdx.x; i < n; i += BLK) s += partials[i];
    s = wave_reduce_add(s);
    const int wave = threadIdx.x >> 5, lane = threadIdx.x & 31;
    if (lane == 0) wsum[wave] = s;
    __syncthreads();
    if (threadIdx.x < WAVES) {
        float t = wsum[threadIdx.x];
        #pragma unroll
        for (int m = WAVES / 2; m >= 1; m >>= 1) t += __shfl_xor(t, m, 32);
        if (threadIdx.x == 0) out[0] = t * inv_batch;
    }
}

extern "C" void kernel_launch(void* const* d_in, const int* in_sizes, int n_in,
                              void* d_out, int out_size, void* d_ws, size_t ws_size,
                              hipStream_t stream) {
    const float* preds  = (const float*)d_in[0];
    const float* labels = (const float*)d_in[1];
    float* out  = (float*)d_out;
    float* part = (float*)d_ws;

    const int total_floats = in_sizes[0];              // 16384*7*7*30
    const int ncells = total_floats / CELL_FLOATS;     // 802816
    const int nblocks = (ncells + BLK - 1) / BLK;      // 3136
    const int batch = ncells / 49;                     // 16384

    yolo_loss_partial<<<nblocks, BLK, 0, stream>>>(preds, labels, part, ncells);
    yolo_loss_final<<<1, BLK, 0, stream>>>(part, nblocks, out, 1.0f / (float)batch);
}